// ODEBlock_65403761983978
// MI455X (gfx1250) — compile-verified
//
#include <hip/hip_runtime.h>
#include <hip/hip_bf16.h>
#include <math.h>

// Problem constants (match reference)
#define NN 20000
#define EE 640000
#define DD 64
#define NSTEPS 10

typedef __attribute__((ext_vector_type(2))) float v2f;
typedef __attribute__((ext_vector_type(8))) float v8f;

// ---------------------------------------------------------------------------
// h = q @ W  via V_WMMA_F32_16X16X4_F32.  One wave computes one 16x16 tile.
// A 16x4 f32 layout (ISA 7.12.2): lanes0-15 M=0..15 {v0=K0,v1=K1}; lanes16-31 {v0=K2,v1=K3}
// B 4x16: lanes0-15 N=0..15 {v0=K0,v1=K1}; lanes16-31 {v0=K2,v1=K3}
// C/D 16x16 f32: VGPR r -> row r (lanes0-15) / row r+8 (lanes16-31), col = lane%16
// ---------------------------------------------------------------------------
__global__ void gemm_qW_wmma(const float* __restrict__ q, const float* __restrict__ W,
                             float* __restrict__ h) {
    int gw   = (blockIdx.x * blockDim.x + threadIdx.x) >> 5; // global wave id
    int lane = threadIdx.x & 31;
    int rt = gw >> 2;          // row tile (N/16 = 1250)
    int ct = gw & 3;           // col tile (64/16 = 4)
    int i0 = rt * 16, j0 = ct * 16;
    int half = lane >> 4, lr = lane & 15;

    v8f c = {};
    #pragma unroll
    for (int kk = 0; kk < DD; kk += 4) {
        int ka = kk + 2 * half;
        v2f a, b;
        a[0] = q[(size_t)(i0 + lr) * DD + ka];
        a[1] = q[(size_t)(i0 + lr) * DD + ka + 1];
        b[0] = W[(size_t)ka * DD + j0 + lr];
        b[1] = W[(size_t)(ka + 1) * DD + j0 + lr];
        c = __builtin_amdgcn_wmma_f32_16x16x4_f32(false, a, false, b,
                                                  (short)0, c, false, false);
    }
    #pragma unroll
    for (int r = 0; r < 8; ++r)
        h[(size_t)(i0 + r + 8 * half) * DD + j0 + lr] = c[r];
}

// hs[n] = h[n,:].a_src ; hd[n] = h[n,:].a_dst  (one wave per node, 2 elems/lane)
__global__ void node_dots(const float* __restrict__ h, const float* __restrict__ a_src,
                          const float* __restrict__ a_dst, float* __restrict__ hs,
                          float* __restrict__ hd) {
    int wid  = (blockIdx.x * blockDim.x + threadIdx.x) >> 5;
    int lane = threadIdx.x & 31;
    if (wid >= NN) return;
    float x0 = h[(size_t)wid * DD + lane];
    float x1 = h[(size_t)wid * DD + 32 + lane];
    float s = x0 * a_src[lane] + x1 * a_src[32 + lane];
    float d = x0 * a_dst[lane] + x1 * a_dst[32 + lane];
    #pragma unroll
    for (int off = 16; off; off >>= 1) {
        s += __shfl_xor(s, off);
        d += __shfl_xor(d, off);
    }
    if (lane == 0) { hs[wid] = s; hd[wid] = d; }
}

__global__ void edge_scores(const int* __restrict__ ei, const float* __restrict__ hs,
                            const float* __restrict__ hd, float* __restrict__ scores) {
    int e = blockIdx.x * blockDim.x + threadIdx.x;
    if (e >= EE) return;
    float v = hs[ei[e]] + hd[ei[EE + e]];
    scores[e] = v > 0.f ? v : 0.2f * v;     // leaky_relu(0.2)
}

__global__ void zero_ints(int* __restrict__ a, int n) {
    int i = blockIdx.x * blockDim.x + threadIdx.x;
    if (i < n) a[i] = 0;
}

__global__ void count_degrees(const int* __restrict__ ei, int* __restrict__ cnt_d,
                              int* __restrict__ cnt_s) {
    int e = blockIdx.x * blockDim.x + threadIdx.x;
    if (e >= EE) return;
    atomicAdd(&cnt_d[ei[EE + e]], 1);   // in-degree (group by dst)
    atomicAdd(&cnt_s[ei[e]], 1);        // out-degree (group by src)
}

// Single-workgroup exclusive scan over both count arrays (N=20000, 1024 thr)
__global__ void scan_two(const int* __restrict__ cnt_d, int* __restrict__ ptr_d,
                         int* __restrict__ cur_d, const int* __restrict__ cnt_s,
                         int* __restrict__ ptr_s, int* __restrict__ cur_s) {
    __shared__ int sh[1024];
    __shared__ int carry;
    for (int pass = 0; pass < 2; ++pass) {
        const int* cnt = pass ? cnt_s : cnt_d;
        int* ptr = pass ? ptr_s : ptr_d;
        int* cur = pass ? cur_s : cur_d;
        if (threadIdx.x == 0) carry = 0;
        __syncthreads();
        for (int base = 0; base < NN; base += 1024) {
            int i = base + (int)threadIdx.x;
            int v = (i < NN) ? cnt[i] : 0;
            sh[threadIdx.x] = v;
            __syncthreads();
            #pragma unroll
            for (int off = 1; off < 1024; off <<= 1) {
                int t = (threadIdx.x >= (unsigned)off) ? sh[threadIdx.x - off] : 0;
                __syncthreads();
                sh[threadIdx.x] += t;
                __syncthreads();
            }
            int inc  = sh[threadIdx.x];
            int excl = carry + inc - v;
            if (i < NN) { ptr[i] = excl; cur[i] = excl; }
            __syncthreads();
            if (threadIdx.x == 1023) carry += inc;
            __syncthreads();
        }
        if (threadIdx.x == 0) ptr[NN] = carry;
        __syncthreads();
    }
}

__global__ void scatter_csr(const int* __restrict__ ei, int* __restrict__ cur_d,
                            int* __restrict__ col_d, int* __restrict__ eid_d,
                            int* __restrict__ cur_s, int* __restrict__ col_s,
                            int* __restrict__ eid_s) {
    int e = blockIdx.x * blockDim.x + threadIdx.x;
    if (e >= EE) return;
    int s = ei[e], d = ei[EE + e];
    int pd = atomicAdd(&cur_d[d], 1);
    col_d[pd] = s; eid_d[pd] = e;
    int ps = atomicAdd(&cur_s[s], 1);
    col_s[ps] = d; eid_s[ps] = e;
}

// Deterministic ordering: insertion-sort each node's segment by edge id (once).
__global__ void sort_segments(const int* __restrict__ ptr, int* __restrict__ eid,
                              int* __restrict__ col) {
    int node = blockIdx.x * blockDim.x + threadIdx.x;
    if (node >= NN) return;
    int b = ptr[node], e = ptr[node + 1];
    for (int i = b + 1; i < e; ++i) {
        int ke = eid[i], kc = col[i];
        int j = i - 1;
        while (j >= b && eid[j] > ke) {
            eid[j + 1] = eid[j]; col[j + 1] = col[j]; --j;
        }
        eid[j + 1] = ke; col[j + 1] = kc;
    }
}

// Numerically-stable softmax over each dst node's incoming edges (wave/node).
__global__ void edge_softmax(const int* __restrict__ ptr, const int* __restrict__ eid,
                             const float* __restrict__ scores, float* __restrict__ alpha_sorted,
                             float* __restrict__ alpha_by_edge) {
    int wid  = (blockIdx.x * blockDim.x + threadIdx.x) >> 5;
    int lane = threadIdx.x & 31;
    if (wid >= NN) return;
    int b = ptr[wid], e = ptr[wid + 1];
    float m = -3.402823e38f;
    for (int j = b + lane; j < e; j += 32) m = fmaxf(m, scores[eid[j]]);
    #pragma unroll
    for (int off = 16; off; off >>= 1) m = fmaxf(m, __shfl_xor(m, off));
    float s = 0.f;
    for (int j = b + lane; j < e; j += 32) {
        float ev = expf(scores[eid[j]] - m);
        alpha_sorted[j] = ev;
        s += ev;
    }
    #pragma unroll
    for (int off = 16; off; off >>= 1) s += __shfl_xor(s, off);
    float inv = 1.0f / (s + 1e-16f);
    for (int j = b + lane; j < e; j += 32) {
        float a = alpha_sorted[j] * inv;
        alpha_sorted[j] = a;
        alpha_by_edge[eid[j]] = a;
    }
}

__global__ void gather_alpha(const int* __restrict__ eid_s, const float* __restrict__ alpha,
                             float* __restrict__ alpha_s) {
    int j = blockIdx.x * blockDim.x + threadIdx.x;
    if (j < EE) alpha_s[j] = alpha[eid_s[j]];
}

// CSR SpMM: out[row,:] = scale * sum_j val[j] * x[col[j],:]  (wave/row, float2/lane)
__global__ void spmm_gather(const int* __restrict__ ptr, const int* __restrict__ col,
                            const float* __restrict__ val, const float* __restrict__ x,
                            float* __restrict__ out, float scale) {
    int row  = (blockIdx.x * blockDim.x + threadIdx.x) >> 5;
    int lane = threadIdx.x & 31;
    if (row >= NN) return;
    int b = ptr[row], e = ptr[row + 1];
    float ax = 0.f, ay = 0.f;
    int j = b;
    for (; j + 1 < e; j += 2) {
        if (j + 7 < e) {   // prefetch gathered rows ~3 iterations ahead (L2-resident)
            __builtin_prefetch(x + (size_t)col[j + 6] * DD + (lane << 1), 0, 1);
            __builtin_prefetch(x + (size_t)col[j + 7] * DD + (lane << 1), 0, 1);
        }
        int   c0 = col[j],     c1 = col[j + 1];
        float a0 = val[j],     a1 = val[j + 1];
        const float2 v0 = *(const float2*)(x + (size_t)c0 * DD + (lane << 1));
        const float2 v1 = *(const float2*)(x + (size_t)c1 * DD + (lane << 1));
        ax += a0 * v0.x + a1 * v1.x;
        ay += a0 * v0.y + a1 * v1.y;
    }
    if (j < e) {
        int   c0 = col[j];
        float a0 = val[j];
        const float2 v0 = *(const float2*)(x + (size_t)c0 * DD + (lane << 1));
        ax += a0 * v0.x; ay += a0 * v0.y;
    }
    float2* o = (float2*)(out + (size_t)row * DD + (lane << 1));
    float2 r; r.x = scale * ax; r.y = scale * ay;
    *o = r;
}

// acc = (first ? w*k : acc + w*k);  t = y + c*k
__global__ void stage_update(float* __restrict__ t, const float* __restrict__ y,
                             const float* __restrict__ k, float* __restrict__ acc,
                             float c, float w, int first, int n) {
    int i = blockIdx.x * blockDim.x + threadIdx.x;
    if (i >= n) return;
    float kv = k[i];
    acc[i] = first ? (w * kv) : (acc[i] + w * kv);
    t[i] = y[i] + c * kv;
}

// y += dt6 * (acc + k)
__global__ void final_update(float* __restrict__ y, const float* __restrict__ acc,
                             const float* __restrict__ k, float dt6, int n) {
    int i = blockIdx.x * blockDim.x + threadIdx.x;
    if (i >= n) return;
    y[i] += dt6 * (acc[i] + k[i]);
}

// ---------------------------------------------------------------------------
extern "C" void kernel_launch(void* const* d_in, const int* in_sizes, int n_in,
                              void* d_out, int out_size, void* d_ws, size_t ws_size,
                              hipStream_t stream) {
    const float* q     = (const float*)d_in[0];
    const float* p     = (const float*)d_in[1];
    const int*   ei    = (const int*)d_in[2];     // [2,E]: row0=src, row1=dst
    const float* W     = (const float*)d_in[3];
    const float* a_src = (const float*)d_in[4];
    const float* a_dst = (const float*)d_in[5];

    const size_t ND = (size_t)NN * DD;
    const float DT = 0.1f, DT2 = 0.05f, DT6 = 0.1f / 6.0f;

    // workspace carve-up (256B aligned slices)
    char* base = (char*)d_ws;
    auto alloc = [&](size_t bytes) -> char* {
        char* r = base;
        base += (bytes + 255) & ~(size_t)255;
        return r;
    };
    float* h        = (float*)alloc(ND * 4);
    float* hs       = (float*)alloc(NN * 4);
    float* hd       = (float*)alloc(NN * 4);
    float* scores   = (float*)alloc(EE * 4);
    float* alpha    = (float*)alloc(EE * 4);
    float* alpha_d  = (float*)alloc(EE * 4);
    float* alpha_s  = (float*)alloc(EE * 4);
    float* tq       = (float*)alloc(ND * 4);
    float* tp       = (float*)alloc(ND * 4);
    float* kq       = (float*)alloc(ND * 4);
    float* kp       = (float*)alloc(ND * 4);
    float* aq       = (float*)alloc(ND * 4);
    float* ap       = (float*)alloc(ND * 4);
    int*   cnt_d    = (int*)alloc((size_t)2 * NN * 4);   // cnt_d | cnt_s contiguous
    int*   cnt_s    = cnt_d + NN;
    int*   ptr_d    = (int*)alloc((NN + 1) * 4);
    int*   cur_d    = (int*)alloc(NN * 4);
    int*   ptr_s    = (int*)alloc((NN + 1) * 4);
    int*   cur_s    = (int*)alloc(NN * 4);
    int*   col_d    = (int*)alloc(EE * 4);
    int*   eid_d    = (int*)alloc(EE * 4);
    int*   col_s    = (int*)alloc(EE * 4);
    int*   eid_s    = (int*)alloc(EE * 4);
    (void)ws_size; (void)n_in; (void)in_sizes; (void)out_size;

    const int BLK = 256;
    const int gEdges = (EE + BLK - 1) / BLK;                 // 2500
    const int gNodesWave = (NN * 32 + BLK - 1) / BLK;        // 2500 (wave/node)
    const int gNodes = (NN + BLK - 1) / BLK;                 // 79
    const int gElems = (int)((ND + BLK - 1) / BLK);          // 5000
    const int gGemm  = (NN / 16) * 4 * 32 / BLK;             // 625 blocks, all waves active

    // --- one-time attention setup (runs every call; deterministic) ---
    zero_ints<<<(2 * NN + BLK - 1) / BLK, BLK, 0, stream>>>(cnt_d, 2 * NN);
    gemm_qW_wmma<<<gGemm, BLK, 0, stream>>>(q, W, h);
    node_dots<<<gNodesWave, BLK, 0, stream>>>(h, a_src, a_dst, hs, hd);
    edge_scores<<<gEdges, BLK, 0, stream>>>(ei, hs, hd, scores);
    count_degrees<<<gEdges, BLK, 0, stream>>>(ei, cnt_d, cnt_s);
    scan_two<<<1, 1024, 0, stream>>>(cnt_d, ptr_d, cur_d, cnt_s, ptr_s, cur_s);
    scatter_csr<<<gEdges, BLK, 0, stream>>>(ei, cur_d, col_d, eid_d, cur_s, col_s, eid_s);
    sort_segments<<<gNodes, BLK, 0, stream>>>(ptr_d, eid_d, col_d);
    sort_segments<<<gNodes, BLK, 0, stream>>>(ptr_s, eid_s, col_s);
    edge_softmax<<<gNodesWave, BLK, 0, stream>>>(ptr_d, eid_d, scores, alpha_d, alpha);
    gather_alpha<<<gEdges, BLK, 0, stream>>>(eid_s, alpha, alpha_s);

    // --- state lives in d_out: [qf ; pf] ---
    float* yq = (float*)d_out;
    float* yp = yq + ND;
    hipMemcpyAsync(yq, q, ND * 4, hipMemcpyDeviceToDevice, stream);
    hipMemcpyAsync(yp, p, ND * 4, hipMemcpyDeviceToDevice, stream);

    // --- 10 RK4 steps; f(q,p) = (A p, -A^T q) ---
    for (int s = 0; s < NSTEPS; ++s) {
        // stage 1: k1 = f(y)
        spmm_gather<<<gNodesWave, BLK, 0, stream>>>(ptr_d, col_d, alpha_d, yp, kq,  1.0f);
        spmm_gather<<<gNodesWave, BLK, 0, stream>>>(ptr_s, col_s, alpha_s, yq, kp, -1.0f);
        stage_update<<<gElems, BLK, 0, stream>>>(tq, yq, kq, aq, DT2, 1.0f, 1, (int)ND);
        stage_update<<<gElems, BLK, 0, stream>>>(tp, yp, kp, ap, DT2, 1.0f, 1, (int)ND);
        // stage 2: k2 = f(y + dt/2 k1)
        spmm_gather<<<gNodesWave, BLK, 0, stream>>>(ptr_d, col_d, alpha_d, tp, kq,  1.0f);
        spmm_gather<<<gNodesWave, BLK, 0, stream>>>(ptr_s, col_s, alpha_s, tq, kp, -1.0f);
        stage_update<<<gElems, BLK, 0, stream>>>(tq, yq, kq, aq, DT2, 2.0f, 0, (int)ND);
        stage_update<<<gElems, BLK, 0, stream>>>(tp, yp, kp, ap, DT2, 2.0f, 0, (int)ND);
        // stage 3: k3 = f(y + dt/2 k2)
        spmm_gather<<<gNodesWave, BLK, 0, stream>>>(ptr_d, col_d, alpha_d, tp, kq,  1.0f);
        spmm_gather<<<gNodesWave, BLK, 0, stream>>>(ptr_s, col_s, alpha_s, tq, kp, -1.0f);
        stage_update<<<gElems, BLK, 0, stream>>>(tq, yq, kq, aq, DT, 2.0f, 0, (int)ND);
        stage_update<<<gElems, BLK, 0, stream>>>(tp, yp, kp, ap, DT, 2.0f, 0, (int)ND);
        // stage 4: k4 = f(y + dt k3);  y += dt/6 (k1 + 2k2 + 2k3 + k4)
        spmm_gather<<<gNodesWave, BLK, 0, stream>>>(ptr_d, col_d, alpha_d, tp, kq,  1.0f);
        spmm_gather<<<gNodesWave, BLK, 0, stream>>>(ptr_s, col_s, alpha_s, tq, kp, -1.0f);
        final_update<<<gElems, BLK, 0, stream>>>(yq, aq, kq, DT6, (int)ND);
        final_update<<<gElems, BLK, 0, stream>>>(yp, ap, kp, DT6, (int)ND);
    }
}